// SGCLayer_20340965114307
// MI455X (gfx1250) — compile-verified
//
#include <hip/hip_runtime.h>
#include <math.h>

typedef __attribute__((ext_vector_type(2))) float v2f;
typedef __attribute__((ext_vector_type(8))) float v8f;

#define NN 50000
#define NE 800000
#define DF 64

// ---------------- zero workspace ----------------
__global__ void sgc_zero_kernel(float* __restrict__ p, int n) {
    int i = blockIdx.x * blockDim.x + threadIdx.x;
    if (i < n) p[i] = 0.0f;
}

// ---------------- degree: atomic count of dst ----------------
__global__ void sgc_degree_kernel(const int* __restrict__ dst, float* __restrict__ deg) {
    int e = blockIdx.x * blockDim.x + threadIdx.x;
    if (e < NE) unsafeAtomicAdd(&deg[dst[e]], 1.0f);
}

// ---------------- norm = clip(deg,1)^-0.5 ; norm2 = norm*norm ----------------
__global__ void sgc_norm_kernel(const float* __restrict__ deg,
                                float* __restrict__ nrm, float* __restrict__ nrm2) {
    int i = blockIdx.x * blockDim.x + threadIdx.x;
    if (i < NN) {
        float d = deg[i];
        d = d < 1.0f ? 1.0f : d;
        float n = 1.0f / sqrtf(d);
        nrm[i]  = n;
        nrm2[i] = n * n;
    }
}

// ---------------- fused scale + scatter-add: acc[dst] += hin[src]*scale[src] ----
// 16 threads per edge, float4 chunk each -> per-edge index loads broadcast in-wave,
// 16B vector gathers, native f32 L2 atomics (working set is L2-resident: 12.8MB).
__global__ void sgc_scatter_kernel(const float* __restrict__ hin,
                                   const float* __restrict__ scale,
                                   const int* __restrict__ src,
                                   const int* __restrict__ dst,
                                   float* __restrict__ acc) {
    unsigned gid = blockIdx.x * blockDim.x + threadIdx.x;
    unsigned e = gid >> 4;
    if (e >= NE) return;
    unsigned c = (gid & 15u) * 4u;
    int s = src[e];
    int d = dst[e];
    float4 v = *(const float4*)(hin + (size_t)s * DF + c);
    float sc = scale[s];
    float* a = acc + (size_t)d * DF + c;
    unsafeAtomicAdd(a + 0, v.x * sc);
    unsafeAtomicAdd(a + 1, v.y * sc);
    unsafeAtomicAdd(a + 2, v.z * sc);
    unsafeAtomicAdd(a + 3, v.w * sc);
}

// ---------------- out = (acc * norm) @ W^T via V_WMMA_F32_16X16X4_F32 ----------
// One wave computes a 16(M) x 16(N) f32 tile, K=64 as 16 chained k=4 WMMAs.
// A 16x4 f32 layout (ISA 7.12.2): lanes 0-15 -> M=lane, VGPR0=K0,VGPR1=K1;
//                                 lanes 16-31 -> M=lane-16, VGPR0=K2,VGPR1=K3.
// B 4x16 f32 mirrors (rows striped across lanes per VGPR, +2 row split).
// 4 waves/block cover the 4 N-tiles (D_OUT=64); grid = 50000/16 = 3125 M-tiles.
__global__ void __launch_bounds__(128)
sgc_gemm_wmma_kernel(const float* __restrict__ h, const float* __restrict__ nrm,
                     const float* __restrict__ W, float* __restrict__ out) {
    const int lane = threadIdx.x & 31;
    const int wave = threadIdx.x >> 5;   // N-tile index 0..3
    const int tm   = blockIdx.x;         // M-tile index 0..3124
    const int row  = lane & 15;
    const int koff = (lane < 16) ? 0 : 2;

    const int node = tm * 16 + row;          // A row for this lane
    const int ocol = wave * 16 + row;        // B column (out channel) for this lane
    const float* arow = h + (size_t)node * DF + koff;
    const float* brow = W + (size_t)ocol * DF + koff;   // B[k,n] = W[n,k]
    const float  scal = nrm[node];           // fold final norm[dst] into A

    v8f c = {};
#pragma unroll
    for (int kk = 0; kk < 16; ++kk) {
        float2 av = *(const float2*)(arow + 4 * kk);
        float2 bv = *(const float2*)(brow + 4 * kk);
        v2f a, b;
        a.x = av.x * scal; a.y = av.y * scal;
        b.x = bv.x;        b.y = bv.y;
        c = __builtin_amdgcn_wmma_f32_16x16x4_f32(
                /*neg_a=*/false, a, /*neg_b=*/false, b,
                /*c_mod=*/(short)0, c, /*reuse_a=*/false, /*reuse_b=*/false);
    }

    // D layout: VGPR r -> M = r + (lane<16 ? 0 : 8), N = lane&15
    float* o = out + ((size_t)(tm * 16 + ((lane < 16) ? 0 : 8))) * DF + wave * 16 + row;
#pragma unroll
    for (int r = 0; r < 8; ++r)
        o[(size_t)r * DF] = c[r];
}

extern "C" void kernel_launch(void* const* d_in, const int* in_sizes, int n_in,
                              void* d_out, int out_size, void* d_ws, size_t ws_size,
                              hipStream_t stream) {
    const float* feat = (const float*)d_in[0];   // [50000,64] f32
    const int*   src  = (const int*)d_in[1];     // [800000]
    const int*   dst  = (const int*)d_in[2];     // [800000]
    const float* W    = (const float*)d_in[3];   // [64,64] f32
    float* out = (float*)d_out;                  // [50000,64] f32

    float* ws    = (float*)d_ws;
    float* deg   = ws;                  // 50000 (padded to 50048)
    float* nrm   = ws + 50048;          // 50000
    float* nrm2  = ws + 100096;         // 50000
    float* acc1  = ws + 150144;         // 50000*64
    float* acc2  = acc1 + (size_t)NN * DF;

    // zero deg + norm regions + both accumulators (deterministic per call)
    const int nzero = 150144 + 2 * NN * DF;
    sgc_zero_kernel<<<(nzero + 255) / 256, 256, 0, stream>>>(ws, nzero);

    sgc_degree_kernel<<<(NE + 255) / 256, 256, 0, stream>>>(dst, deg);
    sgc_norm_kernel<<<(NN + 255) / 256, 256, 0, stream>>>(deg, nrm, nrm2);

    // hop 1: acc1 = scatter(feat * norm[src])
    const int sthreads = NE * 16;
    sgc_scatter_kernel<<<(sthreads + 255) / 256, 256, 0, stream>>>(feat, nrm, src, dst, acc1);
    // hop 2: acc2 = scatter(acc1 * norm[src]^2)   (folds h1 = s1*norm and t2 = h1*norm)
    sgc_scatter_kernel<<<(sthreads + 255) / 256, 256, 0, stream>>>(acc1, nrm2, src, dst, acc2);

    // out = (acc2 * norm) @ W^T   -- fp32 WMMA, exact vs reference precision
    sgc_gemm_wmma_kernel<<<NN / 16, 128, 0, stream>>>(acc2, nrm, W, out);
}